// RootsToPolynomial_4071628996842
// MI455X (gfx1250) — compile-verified
//
#include <hip/hip_runtime.h>
#include <stdint.h>

// ---------------------------------------------------------------------------
// RootsToPolynomial: (B=524288, M=64) roots -> (B, 65) monic poly coefficients
// Thread-per-row triangular recurrence (2080 FMA/row, minimal FLOPs).
// Wave-self-contained CDNA5 async Global<->LDS staging: every VMEM instruction
// moves 512 B contiguous per wave; zero __syncthreads in the kernel.
// ---------------------------------------------------------------------------

#define THREADS        64          // 2 wave32 per block
#define ROWS_PER_BLK   64          // one row per thread
#define ROWS_PER_WAVE  32
#define M_ROOTS        64
#define NCOEF          65
#define IN_PAD_B       272u        // input LDS row stride: 68 dwords (16B aligned, 65<=68)
#define IN_REGION_B    (ROWS_PER_BLK * IN_PAD_B)            // 17408 B
#define OUT_REGION_B   (ROWS_PER_BLK * NCOEF * 4)           // 16640 B
#define SMEM_B         (IN_REGION_B + OUT_REGION_B)         // 34048 B (disjoint in/out)

// ---- CDNA5 async LDS<->global ops (ASYNCcnt), inline asm (toolchain-portable)
__device__ __forceinline__ void async_load_b128(uint32_t lds_addr, uint64_t gaddr) {
    asm volatile("global_load_async_to_lds_b128 %0, %1, off"
                 :: "v"(lds_addr), "v"(gaddr) : "memory");
}
__device__ __forceinline__ void async_store_b128(uint64_t gaddr, uint32_t lds_addr) {
    asm volatile("global_store_async_from_lds_b128 %0, %1, off"
                 :: "v"(gaddr), "v"(lds_addr) : "memory");
}
__device__ __forceinline__ void wait_async0() {
    asm volatile("s_wait_asynccnt 0" ::: "memory");
}
__device__ __forceinline__ void wait_ds0() {
    asm volatile("s_wait_dscnt 0" ::: "memory");
}

__global__ __launch_bounds__(THREADS)
void roots_to_poly_kernel(const float* __restrict__ x, float* __restrict__ out) {
    __shared__ __align__(16) unsigned char smem[SMEM_B];

    const uint32_t t    = threadIdx.x;
    const uint32_t w    = t >> 5;          // wave id within block (0..1)
    const uint32_t lane = t & 31u;
    const uint32_t blk  = blockIdx.x;
    const uint32_t lds0 = (uint32_t)(uintptr_t)&smem[0];   // low 32 bits == LDS offset

    // ------------------------------------------------------------------
    // Wave-private input stage: this wave's 32 rows = 512 b128 chunks.
    // Chunk c = lane + 32*i  -> global contiguous 512 B per instruction.
    // LDS dest: row = 32*w + (c>>4), chunk-in-row = c&15 (= lane&15, const).
    // Per-i strides: global +512 B, LDS +2 rows = +544 B.
    // ------------------------------------------------------------------
    {
        uint64_t g = (uint64_t)(uintptr_t)(x + (size_t)blk * (ROWS_PER_BLK * M_ROOTS))
                   + (uint64_t)w * (ROWS_PER_WAVE * M_ROOTS * 4)   // +8192 B per wave
                   + (uint64_t)lane * 16u;
        uint32_t l = lds0 + (ROWS_PER_WAVE * w + (lane >> 4)) * IN_PAD_B
                   + (lane & 15u) * 16u;
#pragma unroll
        for (int i = 0; i < 16; ++i) {
            async_load_b128(l, g);
            g += 512u;            // 32 chunks ahead in global
            l += 2u * IN_PAD_B;   // 2 rows ahead in padded LDS
        }
    }
    wait_async0();   // per-wave: only this wave's loads feed this wave's rows

    // ------------------------------------------------------------------
    // Triangular recurrence, coefficients resident in VGPRs.
    // 2080 v_fma_f32 per lane (VOPD dual-issues ~2 per cycle slot).
    // ------------------------------------------------------------------
    float a[NCOEF];
    a[0] = 1.0f;
    const float4* rowp = (const float4*)(smem + t * IN_PAD_B);  // 16B aligned
#pragma unroll
    for (int c = 0; c < 16; ++c) {
        const float4 r4 = rowp[c];
#pragma unroll
        for (int s = 0; s < 4; ++s) {
            const int   m  = c * 4 + s;
            const float nx = -((s == 0) ? r4.x : (s == 1) ? r4.y : (s == 2) ? r4.z : r4.w);
            a[m + 1] = nx * a[m];
#pragma unroll
            for (int j = m; j >= 1; --j)
                a[j] = fmaf(nx, a[j - 1], a[j]);
        }
    }

    // ------------------------------------------------------------------
    // Write coefficients into the disjoint LDS output mirror:
    // dword index = 65*t + k; 65 == 1 (mod 64) and region base % 256B == 0
    // -> every ds_store_b32 is bank-conflict-free.
    // ------------------------------------------------------------------
    {
        float* o = (float*)(smem + IN_REGION_B) + t * NCOEF;
#pragma unroll
        for (int k = 0; k < NCOEF; ++k) o[k] = a[k];
    }
    wait_ds0();      // DScnt and ASYNCcnt are independent: commit LDS first

    // ------------------------------------------------------------------
    // Wave-private output stream: this wave's 32 rows = 2080 dwords
    // = 520 b128 chunks (8320 B, 16-aligned). 16 chunks/lane + 8 remainder.
    // LDS mirrors global exactly inside the wave's region.
    // ------------------------------------------------------------------
    {
        const uint64_t gwave = (uint64_t)(uintptr_t)(out + (size_t)blk * (ROWS_PER_BLK * NCOEF))
                             + (uint64_t)w * (ROWS_PER_WAVE * NCOEF * 4);   // +8320 B per wave
        const uint32_t lwave = lds0 + IN_REGION_B + w * (ROWS_PER_WAVE * NCOEF * 4);
        uint64_t g = gwave + (uint64_t)lane * 16u;
        uint32_t l = lwave + lane * 16u;
#pragma unroll
        for (int i = 0; i < 16; ++i) {
            async_store_b128(g, l);
            g += 512u;
            l += 512u;
        }
        if (lane < 8u) {                        // chunks 512..519 of this wave
            const uint32_t off = 512u * 16u + lane * 16u;
            async_store_b128(gwave + off, lwave + off);
        }
    }
    wait_async0();   // explicit; s_endpgm also implies wait-idle
}

extern "C" void kernel_launch(void* const* d_in, const int* in_sizes, int n_in,
                              void* d_out, int out_size, void* d_ws, size_t ws_size,
                              hipStream_t stream) {
    (void)n_in; (void)out_size; (void)d_ws; (void)ws_size;
    const float* x   = (const float*)d_in[0];
    float*       out = (float*)d_out;

    const int rows   = in_sizes[0] / M_ROOTS;     // 524288
    const int blocks = rows / ROWS_PER_BLK;       // 8192 (exact for the fixed shape)

    roots_to_poly_kernel<<<blocks, THREADS, 0, stream>>>(x, out);
}